// PathDecoder_52450140619087
// MI455X (gfx1250) — compile-verified
//
#include <hip/hip_runtime.h>
#include <math.h>

// ---------------------------------------------------------------------------
// CDNA5 (gfx1250) fp32 WMMA decoder for PathDecoder reference.
// All GEMMs run on V_WMMA_F32_16X16X4_F32 (wave32, 16x16 tile per wave).
// Round 2: branch-free K loops; 4-tile-per-wave projection GEMM.
// ---------------------------------------------------------------------------

typedef __attribute__((ext_vector_type(2))) float v2f;
typedef __attribute__((ext_vector_type(8))) float v8f;

#define WMMA_F32X4(a, b, c) \
  __builtin_amdgcn_wmma_f32_16x16x4_f32(false, (a), false, (b), (short)0, (c), false, false)

static __device__ __forceinline__ float sigm(float x) { return 1.0f / (1.0f + expf(-x)); }

static constexpr int BB = 256;
static constexpr int LL = 192;
static constexpr int HH = 320;
static constexpr int EE = 128;
static constexpr int VV = 27000;
static constexpr int NG = 4 * HH; // 1280

// ---------------------------------------------------------------------------
// init: h0 = c0 = sum_l ctx / len ; tok = SOS(1).  grid(B), block(H)
// ---------------------------------------------------------------------------
__global__ void k_init(const float* __restrict__ ctxt, const int* __restrict__ lens,
                       float* __restrict__ h, float* __restrict__ c, int* __restrict__ tok) {
  int b = blockIdx.x, hh = threadIdx.x;
  const float* p = ctxt + (size_t)b * LL * HH + hh;
  float s = 0.f;
  for (int l = 0; l < LL; ++l) s += p[(size_t)l * HH];
  float v = s / (float)lens[b];
  h[b * HH + hh] = v;
  c[b * HH + hh] = v;
  if (hh == 0) tok[b] = 1; // SOS
}

// ---------------------------------------------------------------------------
// gates = embedding[tok] @ w_ih.T + h @ w_hh.T + b_ih + b_hh   [B, 4H]
// Two straight-line K loops (emb K=128, then h K=320): no branches in the
// WMMA loop body -> load_b64 x2 + v_wmma per iteration.
// grid(4H/16/4=20, B/16=16), block(128)
// ---------------------------------------------------------------------------
__global__ void k_gates(const float* __restrict__ emb, const int* __restrict__ tok,
                        const float* __restrict__ h,
                        const float* __restrict__ wih, const float* __restrict__ whh,
                        const float* __restrict__ bih, const float* __restrict__ bhh,
                        float* __restrict__ G) {
  int wv = threadIdx.x >> 5, lane = threadIdx.x & 31;
  int n0 = (blockIdx.x * 4 + wv) * 16;
  int m0 = blockIdx.y * 16;
  int half = lane >> 4, q = lane & 15;
  int am = m0 + q;                                   // A-fragment row (lane&15 -> M)
  const float* erow = emb + (size_t)tok[am] * EE;    // gathered embedding row
  const float* hrow = h + (size_t)am * HH;
  const float* wr1 = wih + (size_t)(n0 + q) * EE;    // B-fragment col (lane&15 -> N)
  const float* wr2 = whh + (size_t)(n0 + q) * HH;
  v8f acc = {};
  for (int kk = 0; kk < EE; kk += 4) {               // emb part
    int k0 = kk + half * 2;                          // (half,vreg) -> K
    v2f a, b;
    a.x = erow[k0]; a.y = erow[k0 + 1];
    b.x = wr1[k0];  b.y = wr1[k0 + 1];
    acc = WMMA_F32X4(a, b, acc);
  }
  for (int kk = 0; kk < HH; kk += 4) {               // hidden part
    int k0 = kk + half * 2;
    v2f a, b;
    a.x = hrow[k0]; a.y = hrow[k0 + 1];
    b.x = wr2[k0];  b.y = wr2[k0 + 1];
    acc = WMMA_F32X4(a, b, acc);
  }
  float bias = bih[n0 + q] + bhh[n0 + q];
#pragma unroll
  for (int j = 0; j < 8; ++j)                        // D: vreg j -> row m0+j+8*half
    G[(size_t)(m0 + j + half * 8) * NG + (n0 + q)] = acc[j] + bias;
}

// ---------------------------------------------------------------------------
// LSTM cell elementwise. grid(B), block(H)
// ---------------------------------------------------------------------------
__global__ void k_cell(const float* __restrict__ G, float* __restrict__ h, float* __restrict__ c) {
  int b = blockIdx.x, hh = threadIdx.x;
  const float* g = G + (size_t)b * NG;
  float gi = g[hh], gf = g[HH + hh], gg = g[2 * HH + hh], go = g[3 * HH + hh];
  int idx = b * HH + hh;
  float cn = sigm(gf) * c[idx] + sigm(gi) * tanhf(gg);
  float hn = sigm(go) * tanhf(cn);
  c[idx] = cn;
  h[idx] = hn;
}

// ---------------------------------------------------------------------------
// C[M=256][N] = A[256][K] @ W[N][K]^T, 4 N-tiles (16x64) per wave.
// One A-fragment load feeds 4 WMMAs per K chunk (load:wmma = 1.25).
// Ragged N handled with index clamp + VALU selects (EXEC stays all-1s).
// grid(ceil(N/256), M/16), block(128)
// ---------------------------------------------------------------------------
__global__ void k_gemm_nt4(const float* __restrict__ A, int lda,
                           const float* __restrict__ W, int ldw,
                           float* __restrict__ C, int ldc, int N, int K) {
  int wv = threadIdx.x >> 5, lane = threadIdx.x & 31;
  int g0 = (blockIdx.x * 4 + wv) * 64;               // first column of this wave
  if (g0 >= N) return;                               // wave-uniform exit
  int m0 = blockIdx.y * 16;
  int half = lane >> 4, q = lane & 15;
  const float* ar = A + (size_t)(m0 + q) * lda;
  const float* wr[4];
  int col[4];
  bool ok[4];
#pragma unroll
  for (int t = 0; t < 4; ++t) {
    col[t] = g0 + 16 * t + q;
    ok[t] = col[t] < N;
    int cc = ok[t] ? col[t] : (N - 1);
    wr[t] = W + (size_t)cc * ldw;
    __builtin_prefetch(wr[t], 0, 1);                 // global_prefetch_b8
  }
  v8f acc0 = {}, acc1 = {}, acc2 = {}, acc3 = {};
  for (int kk = 0; kk < K; kk += 4) {
    int k0 = kk + half * 2;
    v2f a;
    a.x = ar[k0]; a.y = ar[k0 + 1];
    v2f b0, b1, b2, b3;
    b0.x = ok[0] ? wr[0][k0] : 0.f;  b0.y = ok[0] ? wr[0][k0 + 1] : 0.f;
    b1.x = ok[1] ? wr[1][k0] : 0.f;  b1.y = ok[1] ? wr[1][k0 + 1] : 0.f;
    b2.x = ok[2] ? wr[2][k0] : 0.f;  b2.y = ok[2] ? wr[2][k0 + 1] : 0.f;
    b3.x = ok[3] ? wr[3][k0] : 0.f;  b3.y = ok[3] ? wr[3][k0 + 1] : 0.f;
    acc0 = WMMA_F32X4(a, b0, acc0);
    acc1 = WMMA_F32X4(a, b1, acc1);
    acc2 = WMMA_F32X4(a, b2, acc2);
    acc3 = WMMA_F32X4(a, b3, acc3);
  }
  v8f* accs[4] = {&acc0, &acc1, &acc2, &acc3};
#pragma unroll
  for (int t = 0; t < 4; ++t) {
    if (ok[t]) {
#pragma unroll
      for (int j = 0; j < 8; ++j)
        C[(size_t)(m0 + j + half * 8) * ldc + col[t]] = (*accs[t])[j];
    }
  }
}

// ---------------------------------------------------------------------------
// CAT[256][320] = [h | ctx] @ concat_w[320][640]^T  (K=640). Two clean loops.
// grid(5, 16), block(128)
// ---------------------------------------------------------------------------
__global__ void k_gemm_cat(const float* __restrict__ A1, const float* __restrict__ A2,
                           const float* __restrict__ W, float* __restrict__ C) {
  int wv = threadIdx.x >> 5, lane = threadIdx.x & 31;
  int n0 = (blockIdx.x * 4 + wv) * 16;
  int m0 = blockIdx.y * 16;
  int half = lane >> 4, q = lane & 15;
  const float* a1 = A1 + (size_t)(m0 + q) * HH;
  const float* a2 = A2 + (size_t)(m0 + q) * HH;
  const float* wr = W + (size_t)(n0 + q) * (2 * HH); // row of concat_w, K contiguous
  v8f acc = {};
  for (int kk = 0; kk < HH; kk += 4) {               // h part (K 0..319)
    int k0 = kk + half * 2;
    v2f a, b;
    a.x = a1[k0]; a.y = a1[k0 + 1];
    b.x = wr[k0]; b.y = wr[k0 + 1];
    acc = WMMA_F32X4(a, b, acc);
  }
  const float* wrh = wr + HH;                        // second half of the row
  for (int kk = 0; kk < HH; kk += 4) {               // ctx part (K 320..639)
    int k0 = kk + half * 2;
    v2f a, b;
    a.x = a2[k0];  a.y = a2[k0 + 1];
    b.x = wrh[k0]; b.y = wrh[k0 + 1];
    acc = WMMA_F32X4(a, b, acc);
  }
#pragma unroll
  for (int j = 0; j < 8; ++j)
    C[(size_t)(m0 + j + half * 8) * HH + (n0 + q)] = acc[j];
}

// ---------------------------------------------------------------------------
// Masked-softmax Luong attention per batch row.
// grid(B), block(320) = 10 wave32s; wave-per-l dots with __shfl_xor.
// ---------------------------------------------------------------------------
__global__ void k_attn(const float* __restrict__ ctxt, const int* __restrict__ lens,
                       const float* __restrict__ U, float* __restrict__ ctxv) {
  __shared__ float Us[HH];
  __shared__ float sc[LL];
  __shared__ float red[512];
  int b = blockIdx.x, tid = threadIdx.x;
  Us[tid] = U[(size_t)b * HH + tid];
  __syncthreads();
  int wv = tid >> 5, lane = tid & 31;
  int len = lens[b];
  const float* cb = ctxt + (size_t)b * LL * HH;
  for (int l = wv; l < LL; l += 10) {
    const float* cp = cb + (size_t)l * HH;
    float s = 0.f;
    for (int j = lane; j < HH; j += 32) s += cp[j] * Us[j];
    for (int off = 16; off > 0; off >>= 1) s += __shfl_xor(s, off, 32);
    if (lane == 0) sc[l] = (l < len) ? s : s - 1e9f;
  }
  __syncthreads();
  red[tid] = (tid < LL) ? sc[tid] : -3.4e38f;
  if (tid < 512 - HH) red[tid + HH] = -3.4e38f;
  __syncthreads();
  for (int s = 256; s > 0; s >>= 1) {
    if (tid < s) red[tid] = fmaxf(red[tid], red[tid + s]);
    __syncthreads();
  }
  float mx = red[0];
  __syncthreads();
  float e = (tid < LL) ? expf(sc[tid] - mx) : 0.f;
  red[tid] = e;
  if (tid < 512 - HH) red[tid + HH] = 0.f;
  __syncthreads();
  for (int s = 256; s > 0; s >>= 1) {
    if (tid < s) red[tid] += red[tid + s];
    __syncthreads();
  }
  float inv = 1.0f / red[0];
  __syncthreads();
  if (tid < LL) sc[tid] = e * inv;
  __syncthreads();
  float acc = 0.f;
  for (int l = 0; l < LL; ++l) acc += sc[l] * cb[(size_t)l * HH + tid];
  ctxv[(size_t)b * HH + tid] = acc;
}

// ---------------------------------------------------------------------------
// LayerNorm(H=320) + tanh. grid(B), block(320)
// ---------------------------------------------------------------------------
__global__ void k_ln_tanh(const float* __restrict__ CAT, const float* __restrict__ gam,
                          const float* __restrict__ bet, float* __restrict__ T) {
  __shared__ float red[512];
  int b = blockIdx.x, tid = threadIdx.x;
  float x = CAT[(size_t)b * HH + tid];
  red[tid] = x;
  if (tid < 512 - HH) red[tid + HH] = 0.f;
  __syncthreads();
  for (int s = 256; s > 0; s >>= 1) {
    if (tid < s) red[tid] += red[tid + s];
    __syncthreads();
  }
  float mu = red[0] * (1.0f / HH);
  __syncthreads();
  float d = x - mu;
  red[tid] = d * d;
  if (tid < 512 - HH) red[tid + HH] = 0.f;
  __syncthreads();
  for (int s = 256; s > 0; s >>= 1) {
    if (tid < s) red[tid] += red[tid + s];
    __syncthreads();
  }
  float var = red[0] * (1.0f / HH);
  float y = d * rsqrtf(var + 1e-5f) * gam[tid] + bet[tid];
  T[(size_t)b * HH + tid] = tanhf(y);
}

// ---------------------------------------------------------------------------
// First-index argmax over V=27000. grid(B), block(256)
// ---------------------------------------------------------------------------
__global__ void k_argmax(const float* __restrict__ out, int* __restrict__ tok) {
  __shared__ float bv[256];
  __shared__ int bi[256];
  int b = blockIdx.x, tid = threadIdx.x;
  const float* row = out + (size_t)b * VV;
  float best = -3.4e38f;
  int bidx = 0;
  for (int v = tid; v < VV; v += 256) {
    float x = row[v];
    if (x > best) { best = x; bidx = v; }  // strictly greater -> first index kept
  }
  bv[tid] = best; bi[tid] = bidx;
  __syncthreads();
  for (int s = 128; s > 0; s >>= 1) {
    if (tid < s) {
      float xo = bv[tid + s]; int io = bi[tid + s];
      if (xo > bv[tid] || (xo == bv[tid] && io < bi[tid])) { bv[tid] = xo; bi[tid] = io; }
    }
    __syncthreads();
  }
  if (tid == 0) tok[b] = bi[0];
}

// ---------------------------------------------------------------------------
extern "C" void kernel_launch(void* const* d_in, const int* in_sizes, int n_in,
                              void* d_out, int out_size, void* d_ws, size_t ws_size,
                              hipStream_t stream) {
  (void)in_sizes; (void)n_in; (void)out_size; (void)ws_size;
  const float* ctxt = (const float*)d_in[0];
  const int*   lens = (const int*)d_in[1];
  const float* emb  = (const float*)d_in[2];
  const float* wih  = (const float*)d_in[3];
  const float* whh  = (const float*)d_in[4];
  const float* bih  = (const float*)d_in[5];
  const float* bhh  = (const float*)d_in[6];
  const float* attw = (const float*)d_in[7];
  const float* catw = (const float*)d_in[8];
  const float* gam  = (const float*)d_in[9];
  const float* bet  = (const float*)d_in[10];
  const float* pw   = (const float*)d_in[11];
  float* out = (float*)d_out;

  float* ws  = (float*)d_ws;
  float* G   = ws;                    // [256][1280]
  float* h   = G + (size_t)BB * NG;   // [256][320]
  float* c   = h + (size_t)BB * HH;
  float* U   = c + (size_t)BB * HH;
  float* cv  = U + (size_t)BB * HH;
  float* CAT = cv + (size_t)BB * HH;
  float* Tv  = CAT + (size_t)BB * HH;
  int*   tok = (int*)(Tv + (size_t)BB * HH);

  const int T = 8; // output_length (setup constant)

  k_init<<<BB, HH, 0, stream>>>(ctxt, lens, h, c, tok);
  for (int t = 0; t < T; ++t) {
    k_gates<<<dim3(NG / 64, BB / 16), 128, 0, stream>>>(emb, tok, h, wih, whh, bih, bhh, G);
    k_cell<<<BB, HH, 0, stream>>>(G, h, c);
    k_gemm_nt4<<<dim3((HH + 255) / 256, BB / 16), 128, 0, stream>>>(h, HH, attw, HH, U, HH, HH, HH);
    k_attn<<<BB, HH, 0, stream>>>(ctxt, lens, U, cv);
    k_gemm_cat<<<dim3(HH / 64, BB / 16), 128, 0, stream>>>(h, cv, catw, CAT);
    k_ln_tanh<<<BB, HH, 0, stream>>>(CAT, gam, bet, Tv);
    float* outt = out + (size_t)t * BB * VV;
    k_gemm_nt4<<<dim3((VV + 255) / 256, BB / 16), 128, 0, stream>>>(Tv, HH, pw, HH, outt, VV, VV, HH);
    k_argmax<<<BB, 256, 0, stream>>>(outt, tok);
  }
}